// ExponentialMovingAverage_64020782514582
// MI455X (gfx1250) — compile-verified
//
#include <hip/hip_runtime.h>
#include <hip/hip_bf16.h>
#include <math.h>

// EMA (adjusted) scan reformulated as chunked affine-scan -> WMMA f32 16x16x4.
// One wave32 per (batch, 16-feature tile): 32 * (256/16) = 512 waves.
// Per 16-step time chunk:  E = M(16x16, lower-tri) x X(16x16) + carry (x) e_last
// executed as 4 chained V_WMMA_F32_16X16X4_F32 ops.
//
// Roofline: 128 MB in + 128 MB out @ 23.3 TB/s ~= 11 us; compute is trivial,
// so the kernel streams every byte once (NT stores), prefetches 8 chunks
// ahead, and keeps the serial carry chain to bpermute + 4 muls + 4 WMMAs.

typedef float v2f __attribute__((ext_vector_type(2)));
typedef float v8f __attribute__((ext_vector_type(8)));

#define EMA_B 32
#define EMA_T 4096
#define EMA_F 256
#define EMA_CH 16
#define EMA_NC (EMA_T / EMA_CH)
#define EMA_ADJ_CHUNKS 16   // chunks where w != 1.0f (t < 216 => first 14; pad to 16)
#define EMA_PF_DIST 8       // prefetch distance in chunks

__device__ __forceinline__ void
ema_build_chunk_matrix(int tau, int row, int hi, float alpha, float oma,
                       v2f am[4], float cr8[8]) {
    // adjust factor active: per-chunk coefficient matrix.
    // A_u = oma/w, Bc_u = alpha/w at t = tau+u;  t==0: e0 = x0 -> A=0, Bc=1.
    float A[16], Bc[16], Q[16];
    float p = powf(oma, (float)(tau + 1));      // oma^(t+1) at u = 0
    for (int u = 0; u < 16; ++u) {
        const int t = tau + u;
        const float w = (t == 0) ? 1.0f : fmaxf(1.0f - p, 1e-10f);
        A[u]  = (t == 0) ? 0.0f : oma / w;
        Bc[u] = (t == 0) ? 1.0f : alpha / w;
        p *= oma;
    }
    Q[0] = 1.0f;
    for (int i = 1; i < 16; ++i) Q[i] = Q[i - 1] * A[i];
#pragma unroll
    for (int k = 0; k < 4; ++k) {
        const int j0 = 4 * k + 2 * hi;
        const int j1 = j0 + 1;
        am[k].x = (j0 <= row) ? Bc[j0] * (Q[row] / Q[j0]) : 0.0f;
        am[k].y = (j1 <= row) ? Bc[j1] * (Q[row] / Q[j1]) : 0.0f;
    }
#pragma unroll
    for (int r = 0; r < 8; ++r) cr8[r] = A[0] * Q[r + 8 * hi];
}

__global__ __launch_bounds__(32) void
ExponentialMovingAverage_64020782514582_kernel(const float* __restrict__ x,
                                               float* __restrict__ out) {
    const float alpha = 2.0f / 26.0f;
    const float oma   = 1.0f - 2.0f / 26.0f;

    const int wave = blockIdx.x;          // 0..511
    const int b    = wave >> 4;           // batch
    const int f0   = (wave & 15) * 16;    // feature tile base

    const int lane = threadIdx.x & 31;
    const int row  = lane & 15;           // M index (A frag) / N col (B,C,D frags)
    const int hi   = lane >> 4;           // half-wave select
    const int col  = row;

    const float* __restrict__ xb = x   + (size_t)b * EMA_T * EMA_F + f0;
    float* __restrict__       ob = out + (size_t)b * EMA_T * EMA_F + f0;

    // ---- constant-regime matrix (w == 1):  M[i][j] = alpha*oma^(i-j), carry[i]=oma^(i+1)
    v2f   ca[4];
    float ccarry[8];
#pragma unroll
    for (int k = 0; k < 4; ++k) {
        const int j0 = 4 * k + 2 * hi;
        const int j1 = j0 + 1;
        ca[k].x = (j0 <= row) ? alpha * powf(oma, (float)(row - j0)) : 0.0f;
        ca[k].y = (j1 <= row) ? alpha * powf(oma, (float)(row - j1)) : 0.0f;
    }
#pragma unroll
    for (int r = 0; r < 8; ++r) ccarry[r] = powf(oma, (float)(r + 8 * hi + 1));

    float elast = 0.0f;   // e_{tau-1} for this lane's column (chunk 0: carry coeff is 0)

    // =========== phase 1: adjust factor active (per-chunk matrices) ===========
    for (int c = 0; c < EMA_ADJ_CHUNKS; ++c) {
        const int tau = c * EMA_CH;
        const float* __restrict__ xt = xb + (size_t)tau * EMA_F;

        v2f bfr[4];
#pragma unroll
        for (int k = 0; k < 4; ++k) {
            const int r0 = 4 * k + 2 * hi;
            bfr[k].x = xt[(size_t)r0 * EMA_F + col];
            bfr[k].y = xt[(size_t)(r0 + 1) * EMA_F + col];
        }
        {   // prefetch ahead: two lanes cover each of the 16 rows of the future tile
            const float* xp = xb + (size_t)(tau + EMA_PF_DIST * EMA_CH) * EMA_F;
            __builtin_prefetch(xp + (size_t)row * EMA_F + hi * 8, 0, 1);
        }

        v2f   am[4];
        float cr8[8];
        ema_build_chunk_matrix(tau, row, hi, alpha, oma, am, cr8);

        v8f acc;
#pragma unroll
        for (int r = 0; r < 8; ++r) acc[r] = cr8[r] * elast;
        acc = __builtin_amdgcn_wmma_f32_16x16x4_f32(false, am[0], false, bfr[0],
                                                    (short)0, acc, false, false);
        acc = __builtin_amdgcn_wmma_f32_16x16x4_f32(false, am[1], false, bfr[1],
                                                    (short)0, acc, false, false);
        acc = __builtin_amdgcn_wmma_f32_16x16x4_f32(false, am[2], false, bfr[2],
                                                    (short)0, acc, false, false);
        acc = __builtin_amdgcn_wmma_f32_16x16x4_f32(false, am[3], false, bfr[3],
                                                    (short)0, acc, false, false);

        float* __restrict__ ot = ob + (size_t)tau * EMA_F;
#pragma unroll
        for (int r = 0; r < 8; ++r)
            __builtin_nontemporal_store(acc[r], &ot[(size_t)(r + 8 * hi) * EMA_F + col]);

        elast = __shfl(acc[7], 16 + col, 32);
    }

    // =========== phase 2: steady state, constant matrix =========
    // prefetch guarded: skipped in the last PF_DIST chunks (c >= NC - PF_DIST)
#pragma unroll 1
    for (int c = EMA_ADJ_CHUNKS; c < EMA_NC; ++c) {
        const int tau = c * EMA_CH;
        const float* __restrict__ xt = xb + (size_t)tau * EMA_F;

        v2f bfr[4];
#pragma unroll
        for (int k = 0; k < 4; ++k) {
            const int r0 = 4 * k + 2 * hi;
            bfr[k].x = xt[(size_t)r0 * EMA_F + col];
            bfr[k].y = xt[(size_t)(r0 + 1) * EMA_F + col];
        }
        if (c < EMA_NC - EMA_PF_DIST) {
            const float* xp = xb + (size_t)(tau + EMA_PF_DIST * EMA_CH) * EMA_F;
            __builtin_prefetch(xp + (size_t)row * EMA_F + hi * 8, 0, 1);
        }

        v8f acc;
#pragma unroll
        for (int r = 0; r < 8; ++r) acc[r] = ccarry[r] * elast;
        acc = __builtin_amdgcn_wmma_f32_16x16x4_f32(false, ca[0], false, bfr[0],
                                                    (short)0, acc, false, false);
        acc = __builtin_amdgcn_wmma_f32_16x16x4_f32(false, ca[1], false, bfr[1],
                                                    (short)0, acc, false, false);
        acc = __builtin_amdgcn_wmma_f32_16x16x4_f32(false, ca[2], false, bfr[2],
                                                    (short)0, acc, false, false);
        acc = __builtin_amdgcn_wmma_f32_16x16x4_f32(false, ca[3], false, bfr[3],
                                                    (short)0, acc, false, false);

        float* __restrict__ ot = ob + (size_t)tau * EMA_F;
#pragma unroll
        for (int r = 0; r < 8; ++r)
            __builtin_nontemporal_store(acc[r], &ot[(size_t)(r + 8 * hi) * EMA_F + col]);

        elast = __shfl(acc[7], 16 + col, 32);
    }
}

extern "C" void kernel_launch(void* const* d_in, const int* in_sizes, int n_in,
                              void* d_out, int out_size, void* d_ws, size_t ws_size,
                              hipStream_t stream) {
    (void)in_sizes; (void)n_in; (void)d_ws; (void)ws_size; (void)out_size;
    const float* x = (const float*)d_in[0];
    float* out = (float*)d_out;
    const int nwaves = EMA_B * (EMA_F / 16);   // 512 waves, one per (b, 16-feature tile)
    ExponentialMovingAverage_64020782514582_kernel<<<dim3(nwaves), dim3(32), 0, stream>>>(x, out);
}